// GptOssExperts_4973572129410
// MI455X (gfx1250) — compile-verified
//
#include <hip/hip_runtime.h>

// ---------------------------------------------------------------------------
// GPT-OSS MoE experts, dense-all-experts, FP32 end-to-end via
// V_WMMA_F32_16X16X4_F32 (memory-bound problem: ~450MB traffic vs 825 GFLOP,
// so fp32 matrix cores are free relative to the 23.3 TB/s HBM floor).
// ---------------------------------------------------------------------------

#define GLU_ALPHA 1.702f
#define GLU_LIMIT 7.0f

typedef float v2f __attribute__((ext_vector_type(2)));
typedef float v8f __attribute__((ext_vector_type(8)));

static constexpr int H     = 2048;   // hidden dim
static constexpr int II    = 2048;   // intermediate dim
static constexpr int TWO_I = 4096;
static constexpr int NE    = 8;      // experts
static constexpr int CHUNK = 256;    // tokens per chunk (ws = 8*256*2048*4 = 16.8 MB)
static constexpr int KB    = 16;     // K-block staged in LDS
static constexpr int LDSS  = KB + 2; // padded LDS row stride (keeps b64 alignment, spreads banks)

__device__ __forceinline__ v8f wmma_f32(v2f a, v2f b, v8f c) {
  // (neg_a, A, neg_b, B, c_mod, C, reuse_a, reuse_b)
  return __builtin_amdgcn_wmma_f32_16x16x4_f32(false, a, false, b, (short)0, c, false, false);
}

// ---------------------------------------------------------------------------
// Kernel 1: act[e, mLoc, n] = GLU( x @ W_gu[e] + b_gu[e] )   (per 256-token chunk)
// Block: 256 thr = 8 waves; tile 64(M) x 32(N_act); wave tile 16x16.
// ---------------------------------------------------------------------------
__global__ __launch_bounds__(256)
void k_gateup_glu(const float* __restrict__ x,    // N x H
                  const float* __restrict__ wgu,  // E x H x 2I
                  const float* __restrict__ bgu,  // E x 2I
                  float* __restrict__ act,        // E x CHUNK x I (workspace)
                  int chunkBase)
{
  __shared__ __align__(16) float As[64 * LDSS];

  const int e      = blockIdx.z;
  const int tid    = threadIdx.x;
  const int lane   = tid & 31;
  const int wave   = tid >> 5;
  const int waveM  = wave & 3;        // 0..3
  const int waveN  = wave >> 2;       // 0..1
  const int lmod   = lane & 15;
  const int half   = lane >> 4;       // 0: K pair 0, 1: K pair 2
  const int mBlock = chunkBase + blockIdx.y * 64;
  const int nAct   = blockIdx.x * 32 + waveN * 16 + lmod;  // act column (lane)
  const int aRow   = waveM * 16 + lmod;                    // A fragment row (M)

  const float* W = wgu + (size_t)e * H * TWO_I;

  v8f cg = {};   // gate accumulator (16x16)
  v8f cu = {};   // up accumulator   (16x16)

  for (int kb = 0; kb < H; kb += KB) {
    // ---- cooperative, coalesced A-tile load: 64 rows x 16 K-cols ----
    {
      const int r  = tid >> 2;
      const int c4 = (tid & 3) * 4;
      const float4 v = *(const float4*)(x + (size_t)(mBlock + r) * H + kb + c4);
      float* dst = &As[r * LDSS + c4];
      dst[0] = v.x; dst[1] = v.y; dst[2] = v.z; dst[3] = v.w;
    }
    __syncthreads();

#pragma unroll
    for (int k4 = 0; k4 < KB / 4; ++k4) {
      const int kA = k4 * 4 + half * 2;
      // A fragment: (A[m][k], A[m][k+1]) per ISA 16x4 f32 layout
      const v2f a = *(const v2f*)(&As[aRow * LDSS + kA]);

      // B: interleaved (gate,up) column pair -> one b64 load per K-row
      const int    krow = kb + kA;
      const size_t bcol = (size_t)2 * nAct;
      const v2f p0 = *(const v2f*)(W + (size_t)krow       * TWO_I + bcol);
      const v2f p1 = *(const v2f*)(W + (size_t)(krow + 1) * TWO_I + bcol);
      const v2f bg = { p0.x, p1.x };
      const v2f bu = { p0.y, p1.y };

      cg = wmma_f32(a, bg, cg);
      cu = wmma_f32(a, bu, cu);
    }
    __syncthreads();
  }

  // ---- bias + clamp + GLU, write activation ----
  const float biasG = bgu[(size_t)e * TWO_I + 2 * nAct];
  const float biasU = bgu[(size_t)e * TWO_I + 2 * nAct + 1];
#pragma unroll
  for (int r = 0; r < 8; ++r) {
    float g = cg[r] + biasG;
    float u = cu[r] + biasU;
    g = fminf(g, GLU_LIMIT);
    u = fminf(fmaxf(u, -GLU_LIMIT), GLU_LIMIT);
    const float sig = 1.0f / (1.0f + __expf(-GLU_ALPHA * g));
    const float a   = (u + 1.0f) * (g * sig);
    const int mLoc = blockIdx.y * 64 + waveM * 16 + r + half * 8;  // C layout: M = r + 8*(lane>=16)
    act[((size_t)e * CHUNK + mLoc) * II + nAct] = a;
  }
}

// ---------------------------------------------------------------------------
// Kernel 2: out[m, h] = sum_e rw[m,e] * ( act[e] @ W_down[e] + b_down[e] )
// Expert loop inside the block -> single plain store, no atomics, no init pass.
// ---------------------------------------------------------------------------
__global__ __launch_bounds__(256)
void k_down_combine(const float* __restrict__ act,  // E x CHUNK x I
                    const float* __restrict__ wd,   // E x I x H
                    const float* __restrict__ bd,   // E x H
                    const float* __restrict__ rw,   // N x E
                    float* __restrict__ out,        // N x H
                    int chunkBase)
{
  __shared__ __align__(16) float As[64 * LDSS];

  const int tid   = threadIdx.x;
  const int lane  = tid & 31;
  const int wave  = tid >> 5;
  const int waveM = wave & 3;
  const int waveN = wave >> 2;
  const int lmod  = lane & 15;
  const int half  = lane >> 4;
  const int hCol  = blockIdx.x * 32 + waveN * 16 + lmod;
  const int mLocB = blockIdx.y * 64 + waveM * 16;   // token tile base within chunk
  const int aRow  = waveM * 16 + lmod;

  // init accumulator with routing-weighted down-proj bias
  float acc[8];
#pragma unroll
  for (int r = 0; r < 8; ++r) {
    const int mG = chunkBase + mLocB + r + half * 8;
    float s = 0.0f;
#pragma unroll
    for (int e = 0; e < NE; ++e)
      s += rw[(size_t)mG * NE + e] * bd[(size_t)e * H + hCol];
    acc[r] = s;
  }

  for (int e = 0; e < NE; ++e) {
    const float* A = act + (size_t)e * CHUNK * II;
    const float* W = wd  + (size_t)e * II * H;
    v8f c = {};

    for (int kb = 0; kb < II; kb += KB) {
      {
        const int r  = tid >> 2;
        const int c4 = (tid & 3) * 4;
        const float4 v = *(const float4*)(A + (size_t)(blockIdx.y * 64 + r) * II + kb + c4);
        float* dst = &As[r * LDSS + c4];
        dst[0] = v.x; dst[1] = v.y; dst[2] = v.z; dst[3] = v.w;
      }
      __syncthreads();

#pragma unroll
      for (int k4 = 0; k4 < KB / 4; ++k4) {
        const int kA = k4 * 4 + half * 2;
        const v2f a = *(const v2f*)(&As[aRow * LDSS + kA]);
        const int krow = kb + kA;
        const v2f b = { W[(size_t)krow * H + hCol],
                        W[(size_t)(krow + 1) * H + hCol] };
        c = wmma_f32(a, b, c);
      }
      __syncthreads();
    }

#pragma unroll
    for (int r = 0; r < 8; ++r) {
      const int mG = chunkBase + mLocB + r + half * 8;
      acc[r] += rw[(size_t)mG * NE + e] * c[r];
    }
  }

#pragma unroll
  for (int r = 0; r < 8; ++r) {
    const int mG = chunkBase + mLocB + r + half * 8;
    out[(size_t)mG * H + hCol] = acc[r];
  }
}

// ---------------------------------------------------------------------------
extern "C" void kernel_launch(void* const* d_in, const int* in_sizes, int n_in,
                              void* d_out, int out_size, void* d_ws, size_t ws_size,
                              hipStream_t stream) {
  const float* x   = (const float*)d_in[0];  // hidden_states (B,T,H) fp32
  const float* rw  = (const float*)d_in[1];  // routing_weights (N,E)
  const float* wgu = (const float*)d_in[2];  // gate_up_proj (E,H,2I)
  const float* bgu = (const float*)d_in[3];  // gate_up_proj_bias (E,2I)
  const float* wd  = (const float*)d_in[4];  // down_proj (E,I,H)
  const float* bd  = (const float*)d_in[5];  // down_proj_bias (E,H)
  float* out = (float*)d_out;
  float* act = (float*)d_ws;                 // E * CHUNK * I floats = 16.8 MB

  const int N = in_sizes[0] / H;             // 2048 tokens

  for (int cb = 0; cb < N; cb += CHUNK) {
    dim3 g1(II / 32, CHUNK / 64, NE);        // (64, 4, 8)
    k_gateup_glu<<<g1, 256, 0, stream>>>(x, wgu, bgu, act, cb);
    dim3 g2(H / 32, CHUNK / 64);             // (64, 4)
    k_down_combine<<<g2, 256, 0, stream>>>(act, wd, bd, rw, out, cb);
  }
}